// FeatureLeaner_38594576122400
// MI455X (gfx1250) — compile-verified
//
#include <hip/hip_runtime.h>
#include <math.h>

typedef __attribute__((ext_vector_type(2))) float v2f;
typedef __attribute__((ext_vector_type(8))) float v8f;

#define NHW  47
#define NPOS (NHW * NHW)
#define IMH  192
#define IMW  192
#define CPW  206
#define EPSF 1e-8f

__device__ __forceinline__ float norm_sim_term(float sa, float sb, float dot) {
  // sim = (2 - ||a/ (|a|+eps) - b/(|b|+eps)||) / 2, expanded algebraically
  float na = sqrtf(sa) + EPSF;
  float nb = sqrtf(sb) + EPSF;
  float d2 = sa / (na * na) + sb / (nb * nb) - 2.0f * dot / (na * nb);
  d2 = fmaxf(d2, 0.0f);
  return (2.0f - sqrtf(d2)) * 0.5f;
}

// One wave32 per patch position. A-matrix rows = 16 candidates of a group,
// K = 4-chunk of the patch vector; B has all 16 columns equal to the content
// chunk, so every D column holds dot(candidate, content). ||b||^2 is done with
// lane-local FMAs + one xor-16 combine.
__global__ __launch_bounds__(256) void sim_argmax_kernel(
    const float* __restrict__ cont,
    const float* __restrict__ msk,
    const float* __restrict__ cond,
    const float* __restrict__ scont,
    const float* __restrict__ scond,
    float* __restrict__ wsv,
    int*   __restrict__ wsi)
{
  const int wave = threadIdx.x >> 5;
  const int lane = threadIdx.x & 31;
  const int p = blockIdx.x * 8 + wave;
  if (p >= NPOS) return;                 // wave-uniform exit: EXEC all-ones at WMMAs

  const int r = p / NHW, s = p % NHW;
  const int py0 = r * 4, px0 = s * 4;

  const int j0  = (lane >> 4) * 2;       // this half-wave covers K offsets {0,1} or {2,3}
  const int m15 = lane & 15;             // A-matrix row = candidate within group
  const int cyb = 2 * (m15 >> 3);        // candidate search-window row offset (partial)
  const int cx  = 2 * (m15 & 7);         // candidate search-window col offset

  const v8f vzero = {0.f, 0.f, 0.f, 0.f, 0.f, 0.f, 0.f, 0.f};
  v8f accC[4], accS[4];
  float sqC[4], sqS[4];
#pragma unroll
  for (int g = 0; g < 4; ++g) { accC[g] = vzero; accS[g] = vzero; sqC[g] = 0.f; sqS[g] = 0.f; }
  float bsaC = 0.f, bsaS = 0.f;

  // ---- color similarity: K = 3*8*8 = 192, 48 chunks of 4 ----
  for (int c = 0; c < 48; ++c) {
    const int k0  = c * 4 + j0;          // even -> k0, k0+1 share a patch row
    const int ch  = k0 >> 6;
    const int rem = k0 & 63;
    const int ey  = rem >> 3;
    const int ex  = rem & 7;
    const int row = py0 + ey;
    const int col = px0 + ex;
    const int co  = (ch * IMH + row) * IMW + col;
    const float a0 = cont[co], a1 = cont[co + 1];
    const float m0 = msk[co],  m1 = msk[co + 1];
    v2f bf; bf.x = a0; bf.y = a1;        // B fragment: every column = content chunk
    bsaC += a0 * a0 + a1 * a1;
#pragma unroll
    for (int g = 0; g < 4; ++g) {
      const int off = (ch * CPW + row + 4 * g + cyb) * CPW + col + cx;
      const float c0 = cond[off], c1 = cond[off + 1];
      const float A0 = m0 * c0, A1 = m1 * c1;        // mask is {0,1}: where(mask>0, cond, 0)
      sqC[g] += A0 * A0 + A1 * A1;
      v2f af; af.x = A0; af.y = A1;      // A fragment: row = candidate g*16+m15
      accC[g] = __builtin_amdgcn_wmma_f32_16x16x4_f32(
          false, af, false, bf, (short)0, accC[g], false, false);
    }
  }

  // ---- semantic similarity: K = 8*8 = 64, 16 chunks of 4 ----
  for (int c = 0; c < 16; ++c) {
    const int k0 = c * 4 + j0;
    const int ey = k0 >> 3;
    const int ex = k0 & 7;
    const int row = py0 + ey;
    const int col = px0 + ex;
    const int co  = row * IMW + col;
    const float a0 = scont[co], a1 = scont[co + 1];
    const float m0 = msk[co],  m1 = msk[co + 1];     // mask channel 0
    v2f bf; bf.x = a0; bf.y = a1;
    bsaS += a0 * a0 + a1 * a1;
#pragma unroll
    for (int g = 0; g < 4; ++g) {
      const int off = (row + 4 * g + cyb) * CPW + col + cx;
      const float c0 = scond[off], c1 = scond[off + 1];
      const float A0 = m0 * c0, A1 = m1 * c1;
      sqS[g] += A0 * A0 + A1 * A1;
      v2f af; af.x = A0; af.y = A1;
      accS[g] = __builtin_amdgcn_wmma_f32_16x16x4_f32(
          false, af, false, bf, (short)0, accS[g], false, false);
    }
  }

  // full ||a||^2: halves hold complementary K offsets
  const float saC = bsaC + __shfl_xor(bsaC, 16, 32);
  const float saS = bsaS + __shfl_xor(bsaS, 16, 32);

  // lanes {0..7, 24..31} hold matched (dot, ||b||^2) pairs by construction
  const bool act = (lane < 8) || (lane >= 24);
  const int hbit = (lane >= 24) ? 1 : 0;
  const int li   = lane & 7;

  float best = -1.0f;                    // sims are >= 0
  int bestIdx = 64;
#pragma unroll
  for (int g = 0; g < 4; ++g) {
    const float sbC = sqC[g] + __shfl_xor(sqC[g], 16, 32);
    const float sbS = sqS[g] + __shfl_xor(sqS[g], 16, 32);
#pragma unroll
    for (int i = 0; i < 8; ++i) {
      if (act && li == i) {
        const float sim = 0.3f * norm_sim_term(saC, sbC, accC[g][i])
                        + 0.7f * norm_sim_term(saS, sbS, accS[g][i]);
        if (sim > best) { best = sim; bestIdx = g * 16 + hbit * 8 + i; }  // strict >: first-max
      }
    }
  }
  // wave-wide argmax, tie-break to smallest index (matches jnp.argmax)
  for (int off = 16; off > 0; off >>= 1) {
    const float ov = __shfl_xor(best, off, 32);
    const int   oi = __shfl_xor(bestIdx, off, 32);
    if (ov > best || (ov == best && oi < bestIdx)) { best = ov; bestIdx = oi; }
  }
  if (lane == 0) {
    wsv[p] = best;
    wsi[p] = (best > 0.0f) ? bestIdx : 0;
  }
}

// One block per slice (24 slices): mean over growing area, then fill zeros.
__global__ __launch_bounds__(128) void smooth_kernel(const int* __restrict__ wsi,
                                                     int* __restrict__ wss)
{
  __shared__ float ssum[128];
  __shared__ float scnt[128];
  __shared__ float smean;
  const int i = blockIdx.x;              // slice id 0..23 (w=2, swap=1, nW=47)
  const int lo = i;                      // area = [i, min(2i+2,47))
  int hi = 2 * i + 2; if (hi > NHW) hi = NHW;
  const int wdt = hi - lo;
  const int n = NHW * wdt;

  float sum = 0.f, cnt = 0.f;
  for (int e = threadIdx.x; e < n; e += blockDim.x) {
    const int rr = e / wdt, col = lo + e % wdt;
    const float v = (float)wsi[rr * NHW + col];
    sum += v;
    if (v > 0.f) cnt += 1.f;
  }
  ssum[threadIdx.x] = sum; scnt[threadIdx.x] = cnt;
  __syncthreads();
  for (int st = 64; st > 0; st >>= 1) {
    if ((int)threadIdx.x < st) {
      ssum[threadIdx.x] += ssum[threadIdx.x + st];
      scnt[threadIdx.x] += scnt[threadIdx.x + st];
    }
    __syncthreads();
  }
  if (threadIdx.x == 0) smean = rintf(ssum[0] / (scnt[0] + EPSF));  // jnp.round = half-even
  __syncthreads();

  const int slo = 2 * i;
  int shi = 2 * i + 2; if (shi > NHW) shi = NHW;
  const int sw = shi - slo;
  const int m = NHW * sw;
  const int mi = (int)smean;
  for (int e = threadIdx.x; e < m; e += blockDim.x) {
    const int rr = e / sw, col = slo + e % sw;
    const int raw = wsi[rr * NHW + col];
    wss[rr * NHW + col] = (raw > 0) ? raw : mi;
  }
}

// Gather-style fold: fold value at (c,y,x) for patch (r,s) is
// cond[c, y + 2*(k>>3), x + 2*(k&7)]; count is (#covering r)*(#covering s).
__global__ __launch_bounds__(256) void fold_kernel(const float* __restrict__ cond,
                                                   const float* __restrict__ wsv,
                                                   const int* __restrict__ wss,
                                                   float* __restrict__ out)
{
  const int t = blockIdx.x * blockDim.x + threadIdx.x;
  const int NPX = 3 * IMH * IMW;
  if (t >= NPX) return;
  const int x  = t % IMW;
  const int y  = (t / IMW) % IMH;
  const int ch = t / (IMH * IMW);

  const int ty = y - 4;
  int rlo = (ty < 0) ? 0 : (ty >> 2);
  int rhi = y >> 2; if (rhi > NHW - 1) rhi = NHW - 1;
  const int tx = x - 4;
  int slo = (tx < 0) ? 0 : (tx >> 2);
  int shi = x >> 2; if (shi > NHW - 1) shi = NHW - 1;

  float am = 0.f, as = 0.f;
  for (int r = rlo; r <= rhi; ++r) {
    for (int sc = slo; sc <= shi; ++sc) {
      const int q = r * NHW + sc;
      const int k = wss[q];
      am += cond[(ch * CPW + y + 2 * (k >> 3)) * CPW + x + 2 * (k & 7)];
      as += wsv[q];
    }
  }
  const float inv = 1.0f / (float)((rhi - rlo + 1) * (shi - slo + 1));  // exact: 1,2,4
  out[t]       = am * inv;   // mapped
  out[NPX + t] = as * inv;   // simi
}

extern "C" void kernel_launch(void* const* d_in, const int* in_sizes, int n_in,
                              void* d_out, int out_size, void* d_ws, size_t ws_size,
                              hipStream_t stream) {
  (void)in_sizes; (void)n_in; (void)out_size; (void)ws_size;
  const float* content = (const float*)d_in[0];   // (1,3,192,192)
  const float* mask    = (const float*)d_in[1];   // (1,3,192,192)
  const float* cond    = (const float*)d_in[2];   // (1,3,206,206)
  const float* scont   = (const float*)d_in[3];   // (1,1,192,192)
  const float* scond   = (const float*)d_in[4];   // (1,1,206,206)
  float* out = (float*)d_out;                     // mapped ++ simi (2 * 110592 floats)

  float* wsv = (float*)d_ws;                           // 2209 floats: max_val
  int*   wsi = (int*)((char*)d_ws + 8960);             // 2209 ints: max_idx (pre-smooth)
  int*   wss = (int*)((char*)d_ws + 17920);            // 2209 ints: smoothed idx

  sim_argmax_kernel<<<(NPOS + 7) / 8, 256, 0, stream>>>(content, mask, cond, scont, scond, wsv, wsi);
  smooth_kernel<<<24, 128, 0, stream>>>(wsi, wss);
  fold_kernel<<<(3 * IMH * IMW + 255) / 256, 256, 0, stream>>>(cond, wsv, wss, out);
}